// GCN_11527692222479
// MI455X (gfx1250) — compile-verified
//
#include <hip/hip_runtime.h>
#include <hip/hip_bf16.h>

typedef __attribute__((ext_vector_type(2))) float v2f;
typedef __attribute__((ext_vector_type(8))) float v8f;

#define F_IN 512
#define HDIM 16
#define CDIM 5
#define H3   32

// ---------------- degree / norm ----------------
__global__ void k_deg_init(float* __restrict__ deg, int n) {
    int i = blockIdx.x * blockDim.x + threadIdx.x;
    if (i < n) deg[i] = 1.0f;  // self-loop contribution
}

__global__ void k_deg_accum(const int* __restrict__ dst, float* __restrict__ deg, int e) {
    int i = blockIdx.x * blockDim.x + threadIdx.x;
    if (i < e) atomicAdd(&deg[dst[i]], 1.0f);
}

__global__ void k_dinv(float* __restrict__ deg, int n) {
    int i = blockIdx.x * blockDim.x + threadIdx.x;
    if (i < n) deg[i] = rsqrtf(deg[i]);  // deg >= 1 always (self-loops)
}

// ---------------- h = x @ W1 via V_WMMA_F32_16X16X4_F32 ----------------
// One wave computes a 32x16 output block (two stacked 16x16 tiles) so each
// B fragment read from LDS feeds two independent WMMA chains.
// W1 staged in LDS pair-interleaved: lds[(kk>>1)*32 + col*2 + (kk&1)] = W1[kk][col]
// so each B fragment (2 consecutive K for one col) is one contiguous b64.
__global__ void __launch_bounds__(256) k_gemm1(const float* __restrict__ x,
                                               const float* __restrict__ W1,
                                               float* __restrict__ h, int n) {
    __shared__ float ldsB[F_IN * HDIM];  // 32 KB
    int tid = threadIdx.x;
    for (int i = tid; i < F_IN * HDIM; i += 256) {
        int kk = i >> 4, col = i & 15;
        ldsB[(kk >> 1) * 32 + col * 2 + (kk & 1)] = W1[i];
    }
    __syncthreads();

    int wave = tid >> 5, lane = tid & 31;
    int tile = blockIdx.x * 8 + wave;        // 32-row tile index
    int ntiles = (n + 31) >> 5;
    if (tile >= ntiles) return;              // wave-uniform: EXEC stays all-ones for WMMA

    int col  = lane & 15;                    // A: M-row within tile; B/D: N-col
    int koff = (lane >> 4) << 1;             // lanes 0-15 -> K=0,1 ; lanes 16-31 -> K=2,3
    int r0 = tile * 32 + col;       if (r0 >= n) r0 = n - 1;
    int r1 = tile * 32 + 16 + col;  if (r1 >= n) r1 = n - 1;
    const float* xr0 = x + (size_t)r0 * F_IN + koff;
    const float* xr1 = x + (size_t)r1 * F_IN + koff;

    v8f acc0 = {};
    v8f acc1 = {};
    for (int kb = 0; kb < F_IN; kb += 64) {
        __builtin_prefetch(xr0 + kb + 64, 0, 3);   // next K-block, near scope
        __builtin_prefetch(xr1 + kb + 64, 0, 3);
#pragma unroll
        for (int kk = 0; kk < 64; kk += 4) {
            int k = kb + kk;
            v2f a0 = *(const v2f*)(xr0 + k);                                // x[r0][k+koff..+1]
            v2f a1 = *(const v2f*)(xr1 + k);                                // x[r1][k+koff..+1]
            v2f b  = *(const v2f*)(&ldsB[((k + koff) >> 1) * 32 + col * 2]); // W1[k+koff..+1][col]
            acc0 = __builtin_amdgcn_wmma_f32_16x16x4_f32(
                false, a0, false, b, (short)0, acc0, false, false);
            acc1 = __builtin_amdgcn_wmma_f32_16x16x4_f32(
                false, a1, false, b, (short)0, acc1, false, false);
        }
    }

    // D layout: VGPR v, lanes 0-15 -> M=v ; lanes 16-31 -> M=v+8 ; N=lane&15
    int m0 = tile * 32 + ((lane >> 4) << 3);
#pragma unroll
    for (int v = 0; v < 8; ++v) {
        int r = m0 + v;
        if (r < n) h[(size_t)r * HDIM + col] = acc0[v];
    }
    int m1 = m0 + 16;
#pragma unroll
    for (int v = 0; v < 8; ++v) {
        int r = m1 + v;
        if (r < n) h[(size_t)r * HDIM + col] = acc1[v];
    }
}

// ---------------- conv1 aggregation ----------------
__global__ void k_self1(const float* __restrict__ h, const float* __restrict__ dinv,
                        float* __restrict__ out1, int n16) {
    int i = blockIdx.x * blockDim.x + threadIdx.x;  // over n*16
    if (i >= n16) return;
    float di = dinv[i >> 4];
    out1[i] = h[i] * di * di;                       // self-loop term (also initializes out1)
}

__global__ void k_scatter1(const int* __restrict__ src, const int* __restrict__ dst,
                           const float* __restrict__ dinv, const float* __restrict__ h,
                           float* __restrict__ out1, int e16) {
    int i = blockIdx.x * blockDim.x + threadIdx.x;  // over e*16
    if (i >= e16) return;
    int eidx = i >> 4, f = i & 15;
    int s = src[eidx], d = dst[eidx];
    float nrm = dinv[s] * dinv[d];
    atomicAdd(&out1[(size_t)d * HDIM + f], h[(size_t)s * HDIM + f] * nrm);
}

// ---------------- ReLU + bias + (16->5) transform + conv2 self-loop init ----------------
__global__ void k_relu_fc2(const float* __restrict__ out1, const float* __restrict__ b1,
                           const float* __restrict__ W2, const float* __restrict__ dinv,
                           float* __restrict__ h2, float* __restrict__ out2, int n) {
    int i = blockIdx.x * blockDim.x + threadIdx.x;
    if (i >= n) return;
    float a[HDIM];
#pragma unroll
    for (int f = 0; f < HDIM; ++f)
        a[f] = fmaxf(out1[(size_t)i * HDIM + f] + b1[f], 0.0f);
    float di = dinv[i];
    float sl = di * di;
#pragma unroll
    for (int c = 0; c < CDIM; ++c) {
        float s = 0.0f;
#pragma unroll
        for (int f = 0; f < HDIM; ++f) s += a[f] * W2[f * CDIM + c];
        h2[(size_t)i * CDIM + c]   = s;
        out2[(size_t)i * CDIM + c] = s * sl;        // self-loop term (initializes out2)
    }
}

__global__ void k_scatter2(const int* __restrict__ src, const int* __restrict__ dst,
                           const float* __restrict__ dinv, const float* __restrict__ h2,
                           float* __restrict__ out2, int e) {
    int i = blockIdx.x * blockDim.x + threadIdx.x;
    if (i >= e) return;
    int s = src[i], d = dst[i];
    float nrm = dinv[s] * dinv[d];
#pragma unroll
    for (int c = 0; c < CDIM; ++c)
        atomicAdd(&out2[(size_t)d * CDIM + c], h2[(size_t)s * CDIM + c] * nrm);
}

// ---------------- bias + MLP head (5->32->5) + log_softmax ----------------
__global__ void k_head(const float* __restrict__ out2, const float* __restrict__ b2,
                       const float* __restrict__ W3, const float* __restrict__ b3,
                       const float* __restrict__ W4, const float* __restrict__ b4,
                       float* __restrict__ y, int n) {
    int i = blockIdx.x * blockDim.x + threadIdx.x;
    if (i >= n) return;
    float v[CDIM];
#pragma unroll
    for (int c = 0; c < CDIM; ++c) v[c] = out2[(size_t)i * CDIM + c] + b2[c];
    float acc[CDIM];
#pragma unroll
    for (int c = 0; c < CDIM; ++c) acc[c] = b4[c];
#pragma unroll 4
    for (int j = 0; j < H3; ++j) {
        float z = b3[j];
#pragma unroll
        for (int c = 0; c < CDIM; ++c) z += v[c] * W3[c * H3 + j];
        z = fmaxf(z, 0.0f);
#pragma unroll
        for (int c = 0; c < CDIM; ++c) acc[c] += z * W4[j * CDIM + c];
    }
    float m = acc[0];
#pragma unroll
    for (int c = 1; c < CDIM; ++c) m = fmaxf(m, acc[c]);
    float s = 0.0f;
#pragma unroll
    for (int c = 0; c < CDIM; ++c) s += expf(acc[c] - m);
    float lse = logf(s);
#pragma unroll
    for (int c = 0; c < CDIM; ++c) y[(size_t)i * CDIM + c] = acc[c] - m - lse;
}

extern "C" void kernel_launch(void* const* d_in, const int* in_sizes, int n_in,
                              void* d_out, int out_size, void* d_ws, size_t ws_size,
                              hipStream_t stream) {
    const float* x  = (const float*)d_in[0];
    const int*   ei = (const int*)  d_in[1];
    const float* W1 = (const float*)d_in[2];
    const float* b1 = (const float*)d_in[3];
    const float* W2 = (const float*)d_in[4];
    const float* b2 = (const float*)d_in[5];
    const float* W3 = (const float*)d_in[6];
    const float* b3 = (const float*)d_in[7];
    const float* W4 = (const float*)d_in[8];
    const float* b4 = (const float*)d_in[9];
    float* y = (float*)d_out;

    const int n = in_sizes[0] / F_IN;      // 100000
    const int e = in_sizes[1] / 2;         // 6400000
    const int* src = ei;
    const int* dst = ei + e;

    // workspace carve-up (256B aligned)
    char* ws = (char*)d_ws;
    size_t off = 0;
    auto carve = [&](size_t bytes) -> void* {
        void* p = ws + off;
        off = (off + bytes + 255) & ~(size_t)255;
        return p;
    };
    float* dinv = (float*)carve((size_t)n * sizeof(float));
    float* h    = (float*)carve((size_t)n * HDIM * sizeof(float));
    float* out1 = (float*)carve((size_t)n * HDIM * sizeof(float));
    float* h2   = (float*)carve((size_t)n * CDIM * sizeof(float));
    float* out2 = (float*)carve((size_t)n * CDIM * sizeof(float));
    (void)ws_size; (void)n_in; (void)out_size;

    const int B = 256;
    // 1) degrees -> dinv
    k_deg_init <<<(n + B - 1) / B, B, 0, stream>>>(dinv, n);
    k_deg_accum<<<(e + B - 1) / B, B, 0, stream>>>(dst, dinv, e);
    k_dinv     <<<(n + B - 1) / B, B, 0, stream>>>(dinv, n);
    // 2) h = x @ W1  (WMMA, 32-row tile per wave, 8 waves/block)
    int ntiles = (n + 31) / 32;
    k_gemm1    <<<(ntiles + 7) / 8, 256, 0, stream>>>(x, W1, h, n);
    // 3) conv1 aggregation
    int n16 = n * HDIM;
    int e16 = e * HDIM;
    k_self1    <<<(n16 + B - 1) / B, B, 0, stream>>>(h, dinv, out1, n16);
    k_scatter1 <<<(e16 + B - 1) / B, B, 0, stream>>>(src, dst, dinv, h, out1, e16);
    // 4) relu + bias + 16->5 transform, init conv2 aggregation with self-loop
    k_relu_fc2 <<<(n + B - 1) / B, B, 0, stream>>>(out1, b1, W2, dinv, h2, out2, n);
    // 5) conv2 edge scatter
    k_scatter2 <<<(e + B - 1) / B, B, 0, stream>>>(src, dst, dinv, h2, out2, e);
    // 6) head MLP + log_softmax
    k_head     <<<(n + B - 1) / B, B, 0, stream>>>(out2, b2, W3, b3, W4, b4, y, n);
}